// GCN_50276887167405
// MI455X (gfx1250) — compile-verified
//
#include <hip/hip_runtime.h>
#include <hip/hip_bf16.h>
#include <stdint.h>

// ---------------------------------------------------------------------------
// GCN forward (2x GCNConv + BN + ReLU, linear head, per-graph mean pool)
// for MI455X (gfx1250, wave32, WMMA).
//
// GEMMs: V_WMMA_F32_16X16X4_F32 (exact fp32), 128x128 weight staged into LDS
// via global_load_async_to_lds_b128 (ASYNCcnt). Edge propagation: one wave
// per edge, contiguous b128 gather + global_atomic_add_f32 scatter
// (the bandwidth-bound part of this workload).
// ---------------------------------------------------------------------------

typedef __attribute__((ext_vector_type(2))) float v2f;
typedef __attribute__((ext_vector_type(8))) float v8f;

#define DFEAT  128
#define NCLS   10
#define NGRAPH 512
#define BN_EPS 1e-5f

// ---------------- utility fills ----------------
__global__ void fill_f32(float* __restrict__ p, float v, long long n) {
  long long i = (long long)blockIdx.x * blockDim.x + threadIdx.x;
  long long stride = (long long)gridDim.x * blockDim.x;
  for (; i < n; i += stride) p[i] = v;
}

// ---------------- degree / dis ----------------
__global__ void deg_accum(const int* __restrict__ ei, float* __restrict__ deg, int nE) {
  int e = blockIdx.x * blockDim.x + threadIdx.x;
  if (e < nE) atomicAdd(&deg[ei[nE + e]], 1.0f);   // col = dst
}

__global__ void rsqrt_inplace(float* __restrict__ d, int n) {
  int i = blockIdx.x * blockDim.x + threadIdx.x;
  if (i < n) d[i] = rsqrtf(d[i]);                  // deg >= 1 always (self loop)
}

// ---------------- WMMA fp32 GEMM:  Out[n,128] = A[n,128] @ W[128,128] -------
// 8 waves / block, each wave: 16 rows x 128 cols, K stepped by 4.
// FULL=true: block fully in range, no guards anywhere (uniform scalar branch).
template <bool FULL>
__device__ __forceinline__ void gemm_wave(const float* __restrict__ A,
                                          const float* __restrict__ wl,
                                          float* __restrict__ Out,
                                          int nRows, int rowbase, int lane) {
  int m = lane & 15;
  int khalf = (lane >> 4) << 1;                    // 0 (lanes 0-15) or 2 (16-31)
  int row = rowbase + m;
  float amask = 1.0f;
  int lrow = row;
  if (!FULL) {                                     // clamp + mask: no exec churn
    amask = (row < nRows) ? 1.0f : 0.0f;
    lrow = (row < nRows) ? row : (nRows - 1);
  }
  const float* arow = A + (long long)lrow * DFEAT;

  v8f acc[8] = {};
  for (int kt = 0; kt < DFEAT / 4; ++kt) {
    int k = kt * 4 + khalf;
    float2 av = *(const float2*)(arow + k);        // one global_load_b64
    v2f a;
    a.x = av.x * amask;                            // folds away when FULL
    a.y = av.y * amask;
#pragma unroll
    for (int nt = 0; nt < 8; ++nt) {
      int col = nt * 16 + m;
      v2f b;
      b.x = wl[k * DFEAT + col];                   // ds_load (2addr-combined)
      b.y = wl[(k + 1) * DFEAT + col];
      acc[nt] = __builtin_amdgcn_wmma_f32_16x16x4_f32(
          false, a, false, b, (short)0, acc[nt], false, false);
    }
  }

  int mrow = rowbase + ((lane >> 4) << 3);         // C/D: lanes>=16 hold M+8
#pragma unroll
  for (int nt = 0; nt < 8; ++nt) {
    int col = nt * 16 + m;
#pragma unroll
    for (int r = 0; r < 8; ++r) {
      int orow = mrow + r;
      if (FULL || orow < nRows)
        Out[(long long)orow * DFEAT + col] = acc[nt][r];
    }
  }
}

__global__ void __launch_bounds__(256)
gemm128_wmma(const float* __restrict__ A, const float* __restrict__ W,
             float* __restrict__ Out, int nRows) {
  __shared__ float wlds[DFEAT * DFEAT];            // 64 KB of the 320 KB/WGP LDS
  int tid = threadIdx.x;

  // Stage W (64 KB) into LDS with CDNA5 async copies: 4096 b128 transfers,
  // 16 per thread. LDS dest byte address = low 32 bits of the generic pointer
  // (aperture base lives in addr[63:32]).
  {
    uint32_t ldsbase = (uint32_t)(uintptr_t)(&wlds[0]);
    for (int i = tid; i < (DFEAT * DFEAT) / 4; i += 256) {
      uint32_t laddr = ldsbase + (uint32_t)i * 16u;
      uint32_t goff = (uint32_t)i * 16u;
      asm volatile("global_load_async_to_lds_b128 %0, %1, %2"
                   :: "v"(laddr), "v"(goff), "s"(W) : "memory");
    }
    asm volatile("s_wait_asynccnt 0x0" ::: "memory");
  }
  __syncthreads();

  int lane = tid & 31;
  int wave = tid >> 5;
  int rowbase = blockIdx.x * 128 + wave * 16;
  if (blockIdx.x * 128 + 128 <= nRows)             // uniform per block
    gemm_wave<true>(A, wlds, Out, nRows, rowbase, lane);
  else
    gemm_wave<false>(A, wlds, Out, nRows, rowbase, lane);
}

// ---------------- edge scatter: out[dst] += h[src]*dis[src]*dis[dst] -------
__global__ void scatter_edges(const int* __restrict__ ei,
                              const float* __restrict__ h,
                              const float* __restrict__ dis,
                              float* __restrict__ out, int nE) {
  long long gid = (long long)blockIdx.x * blockDim.x + threadIdx.x;
  int lane = threadIdx.x & 31;
  long long e = gid >> 5;                          // one wave per edge
  if (e >= nE) return;
  int s = ei[e];
  int d = ei[nE + e];
  float w = dis[s] * dis[d];
  const float* hs = h + (long long)s * DFEAT;
  float* od = out + (long long)d * DFEAT;
  float4 hv = *(const float4*)(hs + lane * 4);     // wave reads one 512B row
  float* o = od + lane * 4;
  atomicAdd(&o[0], hv.x * w);
  atomicAdd(&o[1], hv.y * w);
  atomicAdd(&o[2], hv.z * w);
  atomicAdd(&o[3], hv.w * w);
}

// ---- self-loop + bias + ReLU, accumulate per-feature sum/sumsq (BN stats) --
__global__ void __launch_bounds__(256)
post_stats(float* __restrict__ prop,               // in: scatter out; out: relu
           const float* __restrict__ h,            // pre-propagation features
           const float* __restrict__ dis,
           const float* __restrict__ bias,
           float* __restrict__ fsum, float* __restrict__ fsq, int n) {
  int f = threadIdx.x & 127;
  int rpb = blockDim.x >> 7;                       // rows per block per iter (2)
  int r0 = blockIdx.x * rpb + (threadIdx.x >> 7);
  float ls = 0.0f, lq = 0.0f;
  for (int i = r0; i < n; i += gridDim.x * rpb) {
    float di = dis[i];
    long long idx = (long long)i * DFEAT + f;
    float v = prop[idx] + di * di * h[idx] + bias[f];
    v = v > 0.0f ? v : 0.0f;                       // ReLU
    prop[idx] = v;
    ls += v; lq += v * v;
  }
  __shared__ float ssum[256], ssq[256];
  ssum[threadIdx.x] = ls; ssq[threadIdx.x] = lq;
  __syncthreads();
  if (threadIdx.x < 128) {
    atomicAdd(&fsum[f], ssum[threadIdx.x] + ssum[threadIdx.x + 128]);
    atomicAdd(&fsq[f],  ssq[threadIdx.x]  + ssq[threadIdx.x + 128]);
  }
}

__global__ void stats_finalize(const float* __restrict__ fsum,
                               const float* __restrict__ fsq,
                               float* __restrict__ mu, float* __restrict__ invstd,
                               int n) {
  int f = threadIdx.x;
  if (f < DFEAT) {
    float m = fsum[f] / (float)n;
    float var = fsq[f] / (float)n - m * m;
    mu[f] = m;
    invstd[f] = rsqrtf(var + BN_EPS);
  }
}

// vectorized BN affine: out = (r - mu) * invstd * g + beta  (float4 lanes)
__global__ void bn_apply(const float4* __restrict__ r,
                         const float* __restrict__ mu, const float* __restrict__ invstd,
                         const float* __restrict__ g, const float* __restrict__ beta,
                         float4* __restrict__ out, long long n4) {
  long long i = (long long)blockIdx.x * blockDim.x + threadIdx.x;
  long long stride = (long long)gridDim.x * blockDim.x;
  for (; i < n4; i += stride) {
    int f = (int)(i & 31) * 4;                     // 32 float4 per 128-feat row
    float4 v = r[i];
    float4 o;
    o.x = (v.x - mu[f + 0]) * invstd[f + 0] * g[f + 0] + beta[f + 0];
    o.y = (v.y - mu[f + 1]) * invstd[f + 1] * g[f + 1] + beta[f + 1];
    o.z = (v.z - mu[f + 2]) * invstd[f + 2] * g[f + 2] + beta[f + 2];
    o.w = (v.w - mu[f + 3]) * invstd[f + 3] * g[f + 3] + beta[f + 3];
    out[i] = o;
  }
}

// ---------------- graph pooling ----------------
__global__ void count_nodes(const int* __restrict__ batch, float* __restrict__ cnt, int n) {
  int i = blockIdx.x * blockDim.x + threadIdx.x;
  if (i < n) atomicAdd(&cnt[batch[i]], 1.0f);
}

__global__ void final_pool(const float* __restrict__ h, const float* __restrict__ Wl,
                           const float* __restrict__ bl, const int* __restrict__ batch,
                           float* __restrict__ psum, int n) {
  long long gid = (long long)blockIdx.x * blockDim.x + threadIdx.x;
  if (gid >= (long long)n * NCLS) return;
  int i = (int)(gid / NCLS);
  int c = (int)(gid % NCLS);
  const float* hi = h + (long long)i * DFEAT;
  float v = bl[c];
#pragma unroll 8
  for (int k = 0; k < DFEAT; ++k) v += hi[k] * Wl[k * NCLS + c];
  atomicAdd(&psum[batch[i] * NCLS + c], v);
}

__global__ void pool_div(const float* __restrict__ psum, const float* __restrict__ cnt,
                         float* __restrict__ out) {
  int i = blockIdx.x * blockDim.x + threadIdx.x;
  if (i < NGRAPH * NCLS) out[i] = psum[i] / fmaxf(cnt[i / NCLS], 1.0f);
}

// ---------------------------------------------------------------------------
extern "C" void kernel_launch(void* const* d_in, const int* in_sizes, int n_in,
                              void* d_out, int out_size, void* d_ws, size_t ws_size,
                              hipStream_t stream) {
  // inputs (setup_inputs order)
  const float* x     = (const float*)d_in[0];
  const int*   ei    = (const int*)  d_in[1];
  const int*   batch = (const int*)  d_in[2];
  const float* W1    = (const float*)d_in[3];
  const float* b1    = (const float*)d_in[4];
  const float* g1    = (const float*)d_in[5];
  const float* be1   = (const float*)d_in[6];
  const float* W2    = (const float*)d_in[7];
  const float* b2    = (const float*)d_in[8];
  const float* g2    = (const float*)d_in[9];
  const float* be2   = (const float*)d_in[10];
  const float* Wl    = (const float*)d_in[11];
  const float* bl    = (const float*)d_in[12];
  float* out = (float*)d_out;

  const int N  = in_sizes[0] / DFEAT;   // 50000
  const int nE = in_sizes[1] / 2;       // 600000
  const long long NF = (long long)N * DFEAT;

  // workspace layout (floats); total ~51.5 MB
  float* ws   = (float*)d_ws;
  float* bufA = ws;                     // NF
  float* bufB = ws + NF;                // NF
  float* dis  = bufB + NF;              // N  (holds deg, then rsqrt in place)
  float* fsum = dis + N;                // 128
  float* fsq  = fsum + DFEAT;           // 128
  float* mu   = fsq + DFEAT;            // 128
  float* istd = mu + DFEAT;             // 128
  float* psum = istd + DFEAT;           // 512*10
  float* pcnt = psum + NGRAPH * NCLS;   // 512

  const int T = 256;
  dim3 blk(T);
  int gN    = (N + T - 1) / T;
  int gE    = (nE + T - 1) / T;
  int gNF   = (int)((NF + T - 1) / T);
  int gNF4  = (int)((NF / 4 + T - 1) / T);
  int gGemm = (N + 127) / 128;
  int gScat = (int)(((long long)nE * 32 + T - 1) / T);
  int gPool = (int)(((long long)N * NCLS + T - 1) / T);

  // degree -> dis
  fill_f32<<<gN, blk, 0, stream>>>(dis, 1.0f, N);          // self loop
  deg_accum<<<gE, blk, 0, stream>>>(ei, dis, nE);
  rsqrt_inplace<<<gN, blk, 0, stream>>>(dis, N);

  // ----- layer 1 -----
  gemm128_wmma<<<gGemm, blk, 0, stream>>>(x, W1, bufA, N); // h1 = X@W1
  fill_f32<<<gNF, blk, 0, stream>>>(bufB, 0.0f, NF);
  scatter_edges<<<gScat, blk, 0, stream>>>(ei, bufA, dis, bufB, nE);
  fill_f32<<<1, blk, 0, stream>>>(fsum, 0.0f, 2 * DFEAT);  // fsum+fsq contiguous
  post_stats<<<1024, blk, 0, stream>>>(bufB, bufA, dis, b1, fsum, fsq, N);
  stats_finalize<<<1, 128, 0, stream>>>(fsum, fsq, mu, istd, N);
  bn_apply<<<gNF4, blk, 0, stream>>>((const float4*)bufB, mu, istd, g1, be1,
                                     (float4*)bufA, NF / 4);

  // ----- layer 2 -----
  gemm128_wmma<<<gGemm, blk, 0, stream>>>(bufA, W2, bufB, N); // h2
  fill_f32<<<gNF, blk, 0, stream>>>(bufA, 0.0f, NF);
  scatter_edges<<<gScat, blk, 0, stream>>>(ei, bufB, dis, bufA, nE);
  fill_f32<<<1, blk, 0, stream>>>(fsum, 0.0f, 2 * DFEAT);
  post_stats<<<1024, blk, 0, stream>>>(bufA, bufB, dis, b2, fsum, fsq, N);
  stats_finalize<<<1, 128, 0, stream>>>(fsum, fsq, mu, istd, N);
  bn_apply<<<gNF4, blk, 0, stream>>>((const float4*)bufA, mu, istd, g2, be2,
                                     (float4*)bufB, NF / 4);

  // ----- head + mean pool -----
  fill_f32<<<1, blk, 0, stream>>>(psum, 0.0f, NGRAPH * NCLS + NGRAPH); // psum+pcnt
  count_nodes<<<gN, blk, 0, stream>>>(batch, pcnt, N);
  final_pool<<<gPool, blk, 0, stream>>>(bufB, Wl, bl, batch, psum, N);
  pool_div<<<(NGRAPH * NCLS + T - 1) / T, blk, 0, stream>>>(psum, pcnt, out);
}